// DGCNN_Centerline_Graph_59596966199697
// MI455X (gfx1250) — compile-verified
//
#include <hip/hip_runtime.h>
#include <stdint.h>

// Problem constants (fixed by the reference).
#define NBATCH   4
#define NPTS     65536      // N per batch (power of two -> shifts/masks)
#define MCOLS    12
#define F_BANK   2048
#define D_FEAT   64
#define MIN_W    1e-10f

#define WAVES    8          // waves per block (256 threads, wave32)
#define TILE     32         // points per wave per tile (one point per lane)
#define ITER     4          // tiles per wave (double-buffered async pipeline)

// out0 = [4,67,65536], out1 = [4,68,65536] concatenated flat
#define OUT0_ELEMS ((size_t)NBATCH * 67 * NPTS)

__device__ __forceinline__ uint32_t lds_byte_addr(const void* p) {
    // Low 32 bits of the flat LDS-aperture address are the LDS byte offset.
    return (uint32_t)(uintptr_t)p;
}

__device__ __forceinline__ void async_pv_row(uint32_t lds_dst, uint64_t gsrc) {
    // Copy 48B (one pv row: 12 floats) global -> LDS via the gfx1250 async
    // LDS-DMA path (tracked by ASYNCcnt). IOFFSET is added to BOTH the LDS
    // and global addresses (CDNA5 ISA), so three B128 ops
    // can share the same base operands.
    asm volatile(
        "global_load_async_to_lds_b128 %0, %1, off\n\t"
        "global_load_async_to_lds_b128 %0, %1, off offset:16\n\t"
        "global_load_async_to_lds_b128 %0, %1, off offset:32"
        :: "v"(lds_dst), "v"(gsrc) : "memory");
}

__global__ __launch_bounds__(WAVES * 32) void fuse_gather_kernel(
    const float* __restrict__ pv_o, const float* __restrict__ pv_b,
    const float* __restrict__ feat_o, const float* __restrict__ feat_b,
    float* __restrict__ out)
{
    __shared__ float pvbuf[WAVES][2][TILE * MCOLS];   // 24 KB

    const int lane = threadIdx.x & 31;
    const int wv   = threadIdx.x >> 5;
    const int s    = blockIdx.y;                      // 0 = pv_o path, 1 = pv_b path

    const float* __restrict__ pv   = s ? pv_b  : pv_o;
    const float* __restrict__ feat = s ? feat_b : feat_o;
    const int    C = 67 + s;                          // output channels
    float* __restrict__ outb = out + (size_t)s * OUT0_ELEMS;

    // Contiguous run of ITER*TILE points per wave.
    const long wbase = ((long)blockIdx.x * WAVES + wv) * (long)(ITER * TILE);

    // Prologue: async-prefetch tile 0 into LDS buffer 0.
    {
        const long g = wbase + lane;
        async_pv_row(lds_byte_addr(&pvbuf[wv][0][lane * MCOLS]),
                     (uint64_t)(uintptr_t)(pv + (size_t)g * MCOLS));
    }

    #pragma unroll 1
    for (int t = 0; t < ITER; ++t) {
        // Issue next tile's async copy, then wait for the current tile only.
        if (t + 1 < ITER) {
            const long gn = wbase + (long)(t + 1) * TILE + lane;
            async_pv_row(lds_byte_addr(&pvbuf[wv][(t + 1) & 1][lane * MCOLS]),
                         (uint64_t)(uintptr_t)(pv + (size_t)gn * MCOLS));
            // Async ops complete in order: <=3 outstanding => tile t landed.
            asm volatile("s_wait_asynccnt 0x3" ::: "memory");
        } else {
            asm volatile("s_wait_asynccnt 0x0" ::: "memory");
        }

        const long g = wbase + (long)t * TILE + lane;
        const int  b = (int)(g >> 16);                // g / NPTS
        const int  n = (int)(g & (NPTS - 1));         // g % NPTS

        const float* __restrict__ row = &pvbuf[wv][t & 1][lane * MCOLS];
        const float c0 = row[0], c1 = row[1], c2 = row[2];
        const float clast = row[11];

        float iv[4];
        const float4* __restrict__ rp[4];
        #pragma unroll
        for (int k = 0; k < 4; ++k) {
            int ix = (int)row[3 + k];                 // truncate toward zero
            ix = ix < 0 ? 0 : (ix > F_BANK - 1 ? F_BANK - 1 : ix);
            float w = row[7 + k];
            w = w < MIN_W ? MIN_W : w;
            iv[k] = 1.0f / w;
            rp[k] = (const float4*)(feat + ((size_t)b * F_BANK + ix) * D_FEAT);
        }

        // Channel-major output: per channel, lanes write consecutive n
        // -> every global_store_b32 is a fully coalesced 128B line.
        // Non-temporal hint: output is write-once, keep it from evicting the
        // L2-resident feature bank. Channel stride = 256KB, so constant
        // channel offsets fold into the 24-bit instruction offset.
        float* __restrict__ op = outb + (size_t)b * C * NPTS + n;
        __builtin_nontemporal_store(c0, op + (size_t)0 * NPTS);
        __builtin_nontemporal_store(c1, op + (size_t)1 * NPTS);
        __builtin_nontemporal_store(c2, op + (size_t)2 * NPTS);
        int cw = 3;
        if (s) { __builtin_nontemporal_store(clast, op + (size_t)3 * NPTS); cw = 4; }
        float* __restrict__ wp = op + (size_t)cw * NPTS;

        // D=64 in 16 chunks of float4: 4 gather loads (L2-resident, regular
        // temporal) + 4 FMAs + 4 coalesced NT channel stores per chunk.
        // Fully unrolled: dc is compile-time constant, so gather-load and
        // store offsets become instruction immediates (kills VALU addr math).
        #pragma unroll
        for (int dc = 0; dc < D_FEAT / 4; ++dc) {
            float a0 = 0.f, a1 = 0.f, a2 = 0.f, a3 = 0.f;
            #pragma unroll
            for (int k = 0; k < 4; ++k) {
                const float4 f = rp[k][dc];
                a0 += iv[k] * f.x;
                a1 += iv[k] * f.y;
                a2 += iv[k] * f.z;
                a3 += iv[k] * f.w;
            }
            float* __restrict__ cp = wp + (size_t)(dc * 4) * NPTS;
            __builtin_nontemporal_store(a0, cp);
            __builtin_nontemporal_store(a1, cp + (size_t)1 * NPTS);
            __builtin_nontemporal_store(a2, cp + (size_t)2 * NPTS);
            __builtin_nontemporal_store(a3, cp + (size_t)3 * NPTS);
        }
    }
}

extern "C" void kernel_launch(void* const* d_in, const int* in_sizes, int n_in,
                              void* d_out, int out_size, void* d_ws, size_t ws_size,
                              hipStream_t stream) {
    (void)in_sizes; (void)n_in; (void)out_size; (void)d_ws; (void)ws_size;
    const float* pv_o   = (const float*)d_in[0];
    const float* pv_b   = (const float*)d_in[1];
    const float* feat_o = (const float*)d_in[2];
    const float* feat_b = (const float*)d_in[3];
    float* out = (float*)d_out;

    // points per block = WAVES*TILE*ITER = 1024; per stream: 4*65536/1024 = 256 blocks
    dim3 grid((NBATCH * NPTS) / (WAVES * TILE * ITER), 2, 1);
    dim3 block(WAVES * 32, 1, 1);
    fuse_gather_kernel<<<grid, block, 0, stream>>>(pv_o, pv_b, feat_o, feat_b, out);
}